// MTLU_continuous_74904229642249
// MI455X (gfx1250) — compile-verified
//
#include <hip/hip_runtime.h>

// ---------------------------------------------------------------------------
// MTLU (multi-bin translated linear unit), CDNA5 / gfx1250.
//
// Roofline: 64Mi f32 in + 64Mi f32 out = 512 MB of HBM traffic vs <1 GFLOP of
// ALU work -> strictly HBM-bound (~22 us floor at 23.3 TB/s). Strategy:
//   * 16-byte non-temporal global loads/stores (streamed data >> 192MB L2)
//   * one channel per block tile -> 20-entry (w,b) LUT in LDS, gathered with
//     a single ds_load_b64 per element
//   * LUT source rows staged via the gfx1250 async global->LDS path
//     (global_load_async_to_lds_b32 + s_wait_asynccnt), exercising ASYNCcnt
//   * per-element index math slimmed to ~7 VALU lane-ops (mul by 10.0f, which
//     IS the correctly-rounded float reciprocal of 0.1f; v_cvt_flr_i32_f32;
//     v_med3_i32 clamp) so VALU time (~14 us) stays under the HBM floor —
//     the IEEE divide expansion (~18 ops/elem) would have been ~35 us and
//     made a bandwidth-bound kernel compute-bound.
// ---------------------------------------------------------------------------

#define BIN_NUM   20
#define HALF_BIN  10
#define FEAT      64
// Elements per block: 256 threads * 4 iters * float4 = 4096. H*W = 65536, so
// every block tile sits inside one (batch, channel) slice: 16 tiles/slice.
#define TILES_PER_SLICE_SHIFT 4   // 65536 / 4096 = 16

typedef float v4f __attribute__((ext_vector_type(4)));
typedef float v2f __attribute__((ext_vector_type(2)));
typedef __attribute__((address_space(3))) float lds_f32;

__global__ __launch_bounds__(256)
void mtlu_kernel(const float* __restrict__ x,
                 const float* __restrict__ mtlu_y,
                 const float* __restrict__ mtlu_y_prev,
                 float* __restrict__ out)
{
    __shared__ float raw[2 * BIN_NUM];   // staged y row | y_prev row
    __shared__ v2f   wb[BIN_NUM];        // interleaved (weight, bias) per bin

    const int tid = (int)threadIdx.x;
    const int c   = (int)((blockIdx.x >> TILES_PER_SLICE_SHIFT) & (FEAT - 1));

    // ---- Stage the two 20-float anchor rows into LDS via the CDNA5 async
    // ---- global->LDS datapath (tracked by ASYNCcnt, not LOADcnt).
    if (tid < 2 * BIN_NUM) {
        const float* src = (tid < BIN_NUM)
            ? (mtlu_y      + c * BIN_NUM + tid)
            : (mtlu_y_prev + c * BIN_NUM + (tid - BIN_NUM));
        unsigned lds_off = (unsigned)(size_t)(lds_f32*)(&raw[tid]);
        asm volatile("global_load_async_to_lds_b32 %0, %1, off"
                     :: "v"(lds_off), "v"(src)
                     : "memory");
    }
    asm volatile("s_wait_asynccnt 0x0" ::: "memory");
    __syncthreads();

    // ---- Build the per-bin affine LUT (w = dy/0.1, b = y - dy*(k-9)),
    // ---- matching the reference arithmetic (block-setup cost: IEEE div ok).
    if (tid < BIN_NUM) {
        float yv = raw[tid];
        float yp = raw[BIN_NUM + tid];
        float dy = yv - yp;
        float w  = dy / 0.1f;                         // slope of bin k
        float iv = (float)(tid - (HALF_BIN - 1));     // index_vals[k] = k - 9
        float b  = yv - dy * iv;                      // intercept of bin k
        v2f p; p.x = w; p.y = b;
        wb[tid] = p;
    }
    __syncthreads();

    // ---- Streaming phase: 4 x float4 per thread, fully coalesced,
    // ---- non-temporal so the 512MB stream does not evict L2.
    const v4f* __restrict__ xv = (const v4f*)x;
    v4f*       __restrict__ ov = (v4f*)out;
    const size_t base = (size_t)blockIdx.x * 1024u + (size_t)tid;

    v4f v[4];
#pragma unroll
    for (int i = 0; i < 4; ++i)
        v[i] = __builtin_nontemporal_load(&xv[base + (size_t)i * 256]);

#pragma unroll
    for (int i = 0; i < 4; ++i) {
        v4f r;
#pragma unroll
        for (int e = 0; e < 4; ++e) {
            float xi = v[i][e];
            // k = clip(floor(x * 10.0f) + 10, 0, 19).
            // 10.0f is the round-to-nearest float of 1.0f/0.1f, so this is
            // the correctly-rounded reciprocal-multiply form of x / 0.1f
            // (<=1 ulp from the reference divide, ~8 VALU ops cheaper).
            int k = __float2int_rd(xi * 10.0f) + HALF_BIN;  // v_cvt_flr_i32_f32
            k = min(max(k, 0), BIN_NUM - 1);                // v_med3_i32
            v2f p = wb[k];                 // single ds_load_b64 gather
            r[e] = fmaf(p.x, xi, p.y);
        }
        __builtin_nontemporal_store(r, &ov[base + (size_t)i * 256]);
    }
}

extern "C" void kernel_launch(void* const* d_in, const int* in_sizes, int n_in,
                              void* d_out, int out_size, void* d_ws, size_t ws_size,
                              hipStream_t stream) {
    const float* x  = (const float*)d_in[0];  // [16,64,256,256] f32
    const float* y  = (const float*)d_in[1];  // [64,20] f32
    const float* yp = (const float*)d_in[2];  // [64,20] f32
    // d_in[3] (index_vals) is the analytic ramp k-9; derived in-kernel.
    float* out = (float*)d_out;

    const int total           = in_sizes[0];       // 67,108,864
    const int elems_per_block = 256 * 4 * 4;       // 4096
    const int blocks          = total / elems_per_block;  // 16384

    mtlu_kernel<<<blocks, 256, 0, stream>>>(x, y, yp, out);
}